// ResBlock_26474178412912
// MI455X (gfx1250) — compile-verified
//
#include <hip/hip_runtime.h>
#include <hip/hip_bf16.h>

typedef __attribute__((ext_vector_type(2))) float v2f;
typedef __attribute__((ext_vector_type(8))) float v8f;

#define BN_EPS 1e-5f

__device__ __forceinline__ float elu_f(float z) {
    return z > 0.0f ? z : (__expf(z) - 1.0f);
}

enum { A_DIRECT = 0, A_CAT3 = 1, A_CAT2G = 2, A_CAT2M = 3 };
enum { O_STORE_STATS = 0, O_STATS_ONLY = 1, O_APPLY_RES = 2 };

// One block = 128 threads (4 waves, wave32) = 64-row x NDIM-col output tile.
// Each wave owns a 16-row stripe and iterates V_WMMA_F32_16X16X4_F32 over K.
// W is staged in LDS pre-paired along K (float2 {W[2p][c],W[2p+1][c]}) so each
// B fragment is one aligned ds_load_b64 straight into the WMMA operand pair.
template<int K, int NDIM, int AM, int OM>
__global__ void __launch_bounds__(128)
gemm_bn_kernel(const float* __restrict__ Xa,
               const float* __restrict__ Xb,
               const float* __restrict__ degv,
               const int*   __restrict__ srcIdx,
               const int*   __restrict__ dstIdx,
               const float* __restrict__ W,      // [K, NDIM] row-major
               const float* __restrict__ bias,   // [NDIM]
               float* __restrict__ Yraw,         // raw y (O_STORE_STATS)
               float* __restrict__ stats,        // [2*NDIM] sum / sumsq
               const float* __restrict__ params, // [2*NDIM] scale / shift
               const float* __restrict__ resid,  // residual (O_APPLY_RES)
               float* __restrict__ Out,          // final output (apply)
               int rows)
{
    constexpr int NT = NDIM / 16;
    constexpr int KP = K / 2;             // number of k-pairs
    __shared__ v2f   Wp[KP * NDIM];       // pre-paired weights
    __shared__ float Al[64][K + 4];       // padded A tile
    __shared__ float sred[2][NDIM];

    const int tid  = threadIdx.x;
    const int row0 = blockIdx.x * 64;

    if constexpr (OM != O_APPLY_RES) {
        for (int i = tid; i < 2 * NDIM; i += 128) (&sred[0][0])[i] = 0.0f;
    }
    for (int i = tid; i < KP * NDIM; i += 128) {
        int p = i / NDIM, c = i % NDIM;
        v2f w;
        w.x = W[(2 * p) * NDIM + c];
        w.y = W[(2 * p + 1) * NDIM + c];
        Wp[i] = w;
    }

    // Cooperative A-tile load (with gather / concat / scatter-mean fused in).
    constexpr int K4 = K / 4;
    for (int idx = tid; idx < 64 * K4; idx += 128) {
        int r    = idx / K4;
        int c4   = (idx % K4) * 4;
        int grow = row0 + r;
        float4 v = make_float4(0.f, 0.f, 0.f, 0.f);
        if (grow < rows) {
            if constexpr (AM == A_DIRECT) {
                v = *(const float4*)(Xa + (size_t)grow * K + c4);
            } else if constexpr (AM == A_CAT3) {
                int seg = c4 >> 5, off = c4 & 31;
                const float* base = (seg == 0) ? (Xa + (size_t)srcIdx[grow] * 32)
                                  : (seg == 1) ? (Xa + (size_t)dstIdx[grow] * 32)
                                               : (Xb + (size_t)grow * 32);
                v = *(const float4*)(base + off);
            } else if constexpr (AM == A_CAT2G) {
                int seg = c4 >> 5, off = c4 & 31;
                const float* base = (seg == 0) ? (Xa + (size_t)srcIdx[grow] * 32)
                                               : (Xb + (size_t)grow * 32);
                v = *(const float4*)(base + off);
            } else { // A_CAT2M: [h_node | msum / max(deg,1)]
                int seg = c4 >> 5, off = c4 & 31;
                if (seg == 0) {
                    v = *(const float4*)(Xa + (size_t)grow * 32 + off);
                } else {
                    float inv = 1.0f / fmaxf(degv[grow], 1.0f);
                    float4 t = *(const float4*)(Xb + (size_t)grow * 32 + off);
                    v = make_float4(t.x * inv, t.y * inv, t.z * inv, t.w * inv);
                }
            }
        }
        Al[r][c4 + 0] = v.x; Al[r][c4 + 1] = v.y;
        Al[r][c4 + 2] = v.z; Al[r][c4 + 3] = v.w;
    }
    __syncthreads();

    const int lane = tid & 31;
    const int wave = tid >> 5;
    const int half = lane >> 4;   // which 16-lane half
    const int l15  = lane & 15;
    const int arow = wave * 16 + l15;

    v8f zero8 = {};
    v8f acc[NT];
    #pragma unroll
    for (int nt = 0; nt < NT; ++nt) acc[nt] = zero8;

    for (int kk = 0; kk < K; kk += 4) {
        // A 16x4 f32 fragment: lanes 0-15 -> K={kk,kk+1}, lanes 16-31 -> K={kk+2,kk+3}
        v2f a = *(const v2f*)&Al[arow][kk + 2 * half];
        const int wbase = ((kk >> 1) + half) * NDIM + l15;
        #pragma unroll
        for (int nt = 0; nt < NT; ++nt) {
            v2f b = Wp[wbase + nt * 16];   // one ds_load_b64, no repack
            acc[nt] = __builtin_amdgcn_wmma_f32_16x16x4_f32(
                false, a, false, b, (short)0, acc[nt], false, false);
        }
    }

    // C/D layout: VGPR i, lanes 0-15 -> M=i, lanes 16-31 -> M=i+8, N=l15.
    const int rbase = row0 + wave * 16 + half * 8;
    if constexpr (OM == O_STORE_STATS || OM == O_STATS_ONLY) {
        #pragma unroll
        for (int nt = 0; nt < NT; ++nt) {
            const int col = nt * 16 + l15;
            const float bv = bias[col];
            float s = 0.0f, s2 = 0.0f;
            #pragma unroll
            for (int i = 0; i < 8; ++i) {
                int grow = rbase + i;
                if (grow < rows) {
                    float y = acc[nt][i] + bv;
                    if constexpr (OM == O_STORE_STATS)
                        Yraw[(size_t)grow * NDIM + col] = y;
                    s += y; s2 += y * y;
                }
            }
            atomicAdd(&sred[0][col], s);
            atomicAdd(&sred[1][col], s2);
        }
        __syncthreads();
        for (int i = tid; i < NDIM; i += 128) {
            atomicAdd(&stats[i],        sred[0][i]);
            atomicAdd(&stats[NDIM + i], sred[1][i]);
        }
    } else { // O_APPLY_RES: recompute path, BN + residual + ELU, direct store
        #pragma unroll
        for (int nt = 0; nt < NT; ++nt) {
            const int col = nt * 16 + l15;
            const float bv = bias[col];
            const float sc = params[col];
            const float sh = params[NDIM + col];
            #pragma unroll
            for (int i = 0; i < 8; ++i) {
                int grow = rbase + i;
                if (grow < rows) {
                    // residual read-once (NT load), output write-once (NT store):
                    // keep the 192MB L2 for the reused [E,32] intermediates.
                    float r = __builtin_nontemporal_load(
                        resid + (size_t)grow * NDIM + col);
                    float z = (acc[nt][i] + bv) * sc + sh + r;
                    __builtin_nontemporal_store(
                        elu_f(z), Out + (size_t)grow * NDIM + col);
                }
            }
        }
    }
}

__global__ void finalize_bn_kernel(const float* __restrict__ stats,
                                   const float* __restrict__ g,
                                   const float* __restrict__ b,
                                   float* __restrict__ params,
                                   int ndim, float invRows)
{
    int c = threadIdx.x + blockIdx.x * blockDim.x;
    if (c < ndim) {
        float mean = stats[c] * invRows;
        float var  = stats[ndim + c] * invRows - mean * mean;
        float rsig = rsqrtf(var + BN_EPS);
        float sc = g[c] * rsig;
        params[c]        = sc;
        params[ndim + c] = b[c] - mean * sc;
    }
}

__global__ void bn_elu_apply32_kernel(const float* __restrict__ raw,
                                      const float* __restrict__ params,
                                      float* __restrict__ out, size_t total)
{
    size_t stride = (size_t)gridDim.x * blockDim.x;
    for (size_t idx = blockIdx.x * (size_t)blockDim.x + threadIdx.x;
         idx < total; idx += stride) {
        int col = (int)(idx & 31);
        float z = raw[idx] * params[col] + params[32 + col];
        out[idx] = elu_f(z);
    }
}

__global__ void bn_elu_scatter_kernel(const float* __restrict__ raw,
                                      const float* __restrict__ params,
                                      const int* __restrict__ dstIdx,
                                      float* __restrict__ msum, size_t total)
{
    size_t stride = (size_t)gridDim.x * blockDim.x;
    for (size_t idx = blockIdx.x * (size_t)blockDim.x + threadIdx.x;
         idx < total; idx += stride) {
        size_t e = idx >> 5;
        int col  = (int)(idx & 31);
        float z = raw[idx] * params[col] + params[32 + col];
        atomicAdd(&msum[(size_t)dstIdx[e] * 32 + col], elu_f(z));
    }
}

__global__ void deg_kernel(const int* __restrict__ dstIdx,
                           float* __restrict__ degb, int E)
{
    int stride = gridDim.x * blockDim.x;
    for (int e = blockIdx.x * blockDim.x + threadIdx.x; e < E; e += stride)
        atomicAdd(&degb[dstIdx[e]], 1.0f);
}

extern "C" void kernel_launch(void* const* d_in, const int* in_sizes, int n_in,
                              void* d_out, int out_size, void* d_ws, size_t ws_size,
                              hipStream_t stream)
{
    (void)in_sizes; (void)n_in; (void)out_size; (void)ws_size;
    const int N = 50000, E = 800000, C = 128;

    const float* node_feats = (const float*)d_in[0];
    const float* edge_feats = (const float*)d_in[1];
    const int*   src        = (const int*)d_in[2];
    const int*   dst        = (const int*)d_in[3];
    const float* pdn_w = (const float*)d_in[4];  const float* pdn_b = (const float*)d_in[5];
    const float* bn1n_g = (const float*)d_in[6]; const float* bn1n_b = (const float*)d_in[7];
    const float* pde_w = (const float*)d_in[8];  const float* pde_b = (const float*)d_in[9];
    const float* bn1e_g = (const float*)d_in[10]; const float* bn1e_b = (const float*)d_in[11];
    const float* e1_w = (const float*)d_in[12];  const float* e1_b = (const float*)d_in[13];
    const float* bne1_g = (const float*)d_in[14]; const float* bne1_b = (const float*)d_in[15];
    const float* e2_w = (const float*)d_in[16];  const float* e2_b = (const float*)d_in[17];
    const float* bne2_g = (const float*)d_in[18]; const float* bne2_b = (const float*)d_in[19];
    const float* nm_w = (const float*)d_in[20];  const float* nm_b = (const float*)d_in[21];
    const float* bnnm_g = (const float*)d_in[22]; const float* bnnm_b = (const float*)d_in[23];
    const float* pun_w = (const float*)d_in[24]; const float* pun_b = (const float*)d_in[25];
    const float* bn2n_g = (const float*)d_in[26]; const float* bn2n_b = (const float*)d_in[27];
    const float* pue_w = (const float*)d_in[28]; const float* pue_b = (const float*)d_in[29];
    const float* bn2e_g = (const float*)d_in[30]; const float* bn2e_b = (const float*)d_in[31];

    // Workspace layout (floats). Zeroed region first (stats, msum, deg).
    float* ws = (float*)d_ws;
    size_t o = 0;
    float* stats_all = ws;            o += 7 * 256;
    float* msum      = ws + o;        o += (size_t)N * 32;
    float* degb      = ws + o;        o += N;
    size_t zeroBytes = o * sizeof(float);
    float* params_all = ws + o;       o += 7 * 256;
    float* rawN   = ws + o;           o += (size_t)N * 32;  // Yn1 raw, reused for ho raw
    float* h_node = ws + o;           o += (size_t)N * 32;
    float* ho     = ws + o;           o += (size_t)N * 32;
    float* rawE   = ws + o;           o += (size_t)E * 32;  // Ye1 -> ex_raw -> m_raw
    float* h_edge = ws + o;           o += (size_t)E * 32;
    float* exb    = ws + o;           o += (size_t)E * 32;

    float* out_node = (float*)d_out;
    float* out_edge = out_node + (size_t)N * C;

    auto st = [&](int i) { return stats_all + i * 256; };
    auto pr = [&](int i) { return params_all + i * 256; };

    hipMemsetAsync(d_ws, 0, zeroBytes, stream);

    dim3 blk(128);
    dim3 gN((N + 63) / 64), gE((E + 63) / 64);

    deg_kernel<<<512, 256, 0, stream>>>(dst, degb, E);

    // Stage 0: h_node = elu(bn(node_feats @ pdn_w + pdn_b))
    gemm_bn_kernel<128, 32, A_DIRECT, O_STORE_STATS><<<gN, blk, 0, stream>>>(
        node_feats, nullptr, nullptr, nullptr, nullptr, pdn_w, pdn_b,
        rawN, st(0), nullptr, nullptr, nullptr, N);
    finalize_bn_kernel<<<1, 128, 0, stream>>>(st(0), bn1n_g, bn1n_b, pr(0), 32, 1.0f / N);
    bn_elu_apply32_kernel<<<2048, 256, 0, stream>>>(rawN, pr(0), h_node, (size_t)N * 32);

    // Stage 1: h_edge = elu(bn(edge_feats @ pde_w + pde_b))
    gemm_bn_kernel<128, 32, A_DIRECT, O_STORE_STATS><<<gE, blk, 0, stream>>>(
        edge_feats, nullptr, nullptr, nullptr, nullptr, pde_w, pde_b,
        rawE, st(1), nullptr, nullptr, nullptr, E);
    finalize_bn_kernel<<<1, 128, 0, stream>>>(st(1), bn1e_g, bn1e_b, pr(1), 32, 1.0f / E);
    bn_elu_apply32_kernel<<<4096, 256, 0, stream>>>(rawE, pr(1), h_edge, (size_t)E * 32);

    // Stage 2: ex = elu(bn([hs|hd|h_edge] @ e1_w + e1_b))  (gather fused)
    gemm_bn_kernel<96, 32, A_CAT3, O_STORE_STATS><<<gE, blk, 0, stream>>>(
        h_node, h_edge, nullptr, src, dst, e1_w, e1_b,
        rawE, st(2), nullptr, nullptr, nullptr, E);
    finalize_bn_kernel<<<1, 128, 0, stream>>>(st(2), bne1_g, bne1_b, pr(2), 32, 1.0f / E);
    bn_elu_apply32_kernel<<<4096, 256, 0, stream>>>(rawE, pr(2), exb, (size_t)E * 32);

    // Stage 3: m = elu(bn([hs|ex] @ e2_w + e2_b)); scatter-sum into msum
    gemm_bn_kernel<64, 32, A_CAT2G, O_STORE_STATS><<<gE, blk, 0, stream>>>(
        h_node, exb, nullptr, src, nullptr, e2_w, e2_b,
        rawE, st(3), nullptr, nullptr, nullptr, E);
    finalize_bn_kernel<<<1, 128, 0, stream>>>(st(3), bne2_g, bne2_b, pr(3), 32, 1.0f / E);
    bn_elu_scatter_kernel<<<4096, 256, 0, stream>>>(rawE, pr(3), dst, msum, (size_t)E * 32);

    // Stage 4: ho = elu(bn([h_node | msum/max(deg,1)] @ nm_w + nm_b))
    gemm_bn_kernel<64, 32, A_CAT2M, O_STORE_STATS><<<gN, blk, 0, stream>>>(
        h_node, msum, degb, nullptr, nullptr, nm_w, nm_b,
        rawN, st(4), nullptr, nullptr, nullptr, N);
    finalize_bn_kernel<<<1, 128, 0, stream>>>(st(4), bnnm_g, bnnm_b, pr(4), 32, 1.0f / N);
    bn_elu_apply32_kernel<<<2048, 256, 0, stream>>>(rawN, pr(4), ho, (size_t)N * 32);

    // Stage 5: node_out = elu(bn(ho @ pun_w + pun_b) + node_feats)  (recompute)
    gemm_bn_kernel<32, 128, A_DIRECT, O_STATS_ONLY><<<gN, blk, 0, stream>>>(
        ho, nullptr, nullptr, nullptr, nullptr, pun_w, pun_b,
        nullptr, st(5), nullptr, nullptr, nullptr, N);
    finalize_bn_kernel<<<1, 128, 0, stream>>>(st(5), bn2n_g, bn2n_b, pr(5), 128, 1.0f / N);
    gemm_bn_kernel<32, 128, A_DIRECT, O_APPLY_RES><<<gN, blk, 0, stream>>>(
        ho, nullptr, nullptr, nullptr, nullptr, pun_w, pun_b,
        nullptr, nullptr, pr(5), node_feats, out_node, N);

    // Stage 6: edge_out = elu(bn(ex @ pue_w + pue_b) + edge_feats)  (recompute)
    gemm_bn_kernel<32, 128, A_DIRECT, O_STATS_ONLY><<<gE, blk, 0, stream>>>(
        exb, nullptr, nullptr, nullptr, nullptr, pue_w, pue_b,
        nullptr, st(6), nullptr, nullptr, nullptr, E);
    finalize_bn_kernel<<<1, 128, 0, stream>>>(st(6), bn2e_g, bn2e_b, pr(6), 128, 1.0f / E);
    gemm_bn_kernel<32, 128, A_DIRECT, O_APPLY_RES><<<gE, blk, 0, stream>>>(
        exb, nullptr, nullptr, nullptr, nullptr, pue_w, pue_b,
        nullptr, nullptr, pr(6), edge_feats, out_edge, E);
}